// HetGraphNet_65128884076772
// MI455X (gfx1250) — compile-verified
//
#include <hip/hip_runtime.h>
#include <hip/hip_bf16.h>

#define NP_N 100000
#define NA_N 100000
#define E_N  3200000

typedef float v2f __attribute__((ext_vector_type(2)));
typedef float v8f __attribute__((ext_vector_type(8)));

static inline int cdiv(int a, int b) { return (a + b - 1) / b; }

__device__ __forceinline__ float lrelu(float a) { return a > 0.f ? a : 0.2f * a; }
__device__ __forceinline__ float reluf(float a) { return a > 0.f ? a : 0.f; }

// order-preserving float atomic max via sign-split int/uint atomics
__device__ __forceinline__ void atomicMaxF(float* addr, float val) {
  if (val >= 0.f) atomicMax((int*)addr, __float_as_int(val));
  else            atomicMin((unsigned int*)addr, __float_as_uint(val));
}

// ---------------------------------------------------------------------------
// WMMA projection: Y[r, 0..ncols) = X[r, 0..K) @ W[K, ncols] + bias
// One wave computes a 16-row tile with v_wmma_f32_16x16x4_f32.
// W is staged zero-padded into LDS, laid out so each lane's (k,k+1) B-pair is
// one contiguous 8-byte ds_load: inner loop = global_load_b64 + ds_load_b64 +
// wmma, with uniform control flow (EXEC all-1s at every WMMA).
// ---------------------------------------------------------------------------
__global__ void wmma_proj_kernel(const float* __restrict__ X,
                                 const float* __restrict__ W,
                                 const float* __restrict__ bias,
                                 float* __restrict__ Y,
                                 int nrows, int K, int ncols, int ldy) {
  // LDS layout: pairs of K-rows, padded to 16 cols:
  //   bsm[((k>>1)*16 + col)*2 + (k&1)] = (col < ncols) ? W[k*ncols+col] : 0
  __shared__ float bsm[128 * 16];  // supports K <= 128
  for (int idx = threadIdx.x; idx < K * 16; idx += blockDim.x) {
    int k = idx >> 4, c = idx & 15;
    bsm[(((k >> 1) << 4) + c) * 2 + (k & 1)] =
        (c < ncols) ? W[(size_t)k * ncols + c] : 0.f;
  }
  __syncthreads();

  int wave = blockIdx.x * (blockDim.x >> 5) + (threadIdx.x >> 5);
  int row0 = wave * 16;
  int lane = threadIdx.x & 31;
  int half = lane >> 4;  // 0 => lanes 0-15 (K pair k0,k0+1), 1 => lanes 16-31 (k0+2,k0+3)
  int l    = lane & 15;

  if (row0 < nrows) {                       // wave-uniform
    const float* xr = X + (size_t)(row0 + l) * K + 2 * half;      // 8B-aligned
    const v2f*   br = (const v2f*)bsm + ((size_t)half * 16 + l);  // pair pointer
    v8f acc = {};
    for (int k0 = 0; k0 < K; k0 += 4) {
      v2f a = *(const v2f*)(xr + k0);       // A 16x4: row M=l, K={k0,k0+1}+2*half
      v2f b = br[(k0 >> 1) * 16];           // B 4x16: col N=l, same K pair
      acc = __builtin_amdgcn_wmma_f32_16x16x4_f32(false, a, false, b, (short)0,
                                                  acc, false, false);
    }
    if (l < ncols) {
      float bv = bias[l];
#pragma unroll
      for (int v = 0; v < 8; ++v) {
        int r = row0 + v + 8 * half;        // C/D layout: VGPR v -> M = v + 8*half
        if (r < nrows) Y[(size_t)r * ldy + l] = acc[v] + bv;
      }
    }
  }
}

// ---------------------------------------------------------------------------
// helpers / node kernels
// ---------------------------------------------------------------------------
__global__ void fill_kernel(float* p, float v, size_t n) {
  size_t i = (size_t)blockIdx.x * blockDim.x + threadIdx.x;
  if (i < n) p[i] = v;
}

// per-node per-head dot with attention vector: out[n*H+h] = sum_d xp[n,h,d]*avec[h,d]
template <int H, int D>
__global__ void alpha_kernel(const float* __restrict__ xp, int xs,
                             const float* __restrict__ avec,
                             float* __restrict__ out, int n) {
  int i = blockIdx.x * blockDim.x + threadIdx.x;
  if (i >= n) return;
#pragma unroll
  for (int h = 0; h < H; ++h) {
    float acc = 0.f;
#pragma unroll
    for (int d = 0; d < D; ++d) acc += xp[(size_t)i * xs + h * D + d] * avec[h * D + d];
    out[(size_t)i * H + h] = acc;
  }
}

// ---------------------------------------------------------------------------
// edge kernels (segment softmax in 3 passes + message scatter)
// ---------------------------------------------------------------------------
template <int H>
__global__ void edge_max_kernel(const int* __restrict__ src, const int* __restrict__ dst,
                                const float* __restrict__ as, const float* __restrict__ ad,
                                float* __restrict__ m, int ne) {
  int e = blockIdx.x * blockDim.x + threadIdx.x;
  if (e >= ne) return;
  int s = src[e], d = dst[e];
#pragma unroll
  for (int h = 0; h < H; ++h) {
    float a = lrelu(as[(size_t)s * H + h] + ad[(size_t)d * H + h]);
    atomicMaxF(&m[(size_t)d * H + h], a);
  }
}

template <int H>
__global__ void edge_sumexp_kernel(const int* __restrict__ src, const int* __restrict__ dst,
                                   const float* __restrict__ as, const float* __restrict__ ad,
                                   const float* __restrict__ m, float* __restrict__ ssum,
                                   int ne) {
  int e = blockIdx.x * blockDim.x + threadIdx.x;
  if (e >= ne) return;
  int s = src[e], d = dst[e];
#pragma unroll
  for (int h = 0; h < H; ++h) {
    float a = lrelu(as[(size_t)s * H + h] + ad[(size_t)d * H + h]);
    atomicAdd(&ssum[(size_t)d * H + h], __expf(a - m[(size_t)d * H + h]));
  }
}

template <int H, int D>
__global__ void edge_msg_kernel(const int* __restrict__ src, const int* __restrict__ dst,
                                const float* __restrict__ as, const float* __restrict__ ad,
                                const float* __restrict__ m, const float* __restrict__ ssum,
                                const float* __restrict__ xp, int xs,
                                float* __restrict__ msg, int ms, int ne) {
  int e = blockIdx.x * blockDim.x + threadIdx.x;
  if (e >= ne) return;
  int s = src[e], d = dst[e];
#pragma unroll
  for (int h = 0; h < H; ++h) {
    float a = lrelu(as[(size_t)s * H + h] + ad[(size_t)d * H + h]);
    float coeff = __expf(a - m[(size_t)d * H + h]) / (ssum[(size_t)d * H + h] + 1e-16f);
#pragma unroll
    for (int dd = 0; dd < D; ++dd)
      atomicAdd(&msg[(size_t)d * ms + h * D + dd],
                coeff * xp[(size_t)s * xs + h * D + dd]);
  }
}

// ---------------------------------------------------------------------------
// semantic attention (_group)
// ---------------------------------------------------------------------------
// partial[f] += sum_n tanh( relu(msg[n]) @ kw[:,f] + kb[f] )
template <int F>
__global__ void sem_reduce_kernel(const float* __restrict__ msg, int ms,
                                  const float* __restrict__ kw,
                                  const float* __restrict__ kb,
                                  float* __restrict__ partial, int n) {
  __shared__ float shacc[F];
  if (threadIdx.x < F) shacc[threadIdx.x] = 0.f;
  __syncthreads();
  int i = blockIdx.x * blockDim.x + threadIdx.x;
  if (i < n) {
    float x[F];
#pragma unroll
    for (int f = 0; f < F; ++f) x[f] = reluf(msg[(size_t)i * ms + f]);
#pragma unroll
    for (int f = 0; f < F; ++f) {
      float acc = kb[f];
#pragma unroll
      for (int g = 0; g < F; ++g) acc += x[g] * kw[g * F + f];
      atomicAdd(&shacc[f], tanhf(acc));
    }
  }
  __syncthreads();
  if (threadIdx.x < F) atomicAdd(&partial[threadIdx.x], shacc[threadIdx.x]);
}

// attn = softmax_t( sum_f q[f] * partial[t*F+f] / n ), T = 2
__global__ void sem_attn_kernel(const float* __restrict__ partial,
                                const float* __restrict__ q, int F, float invn,
                                float* __restrict__ attn) {
  if (threadIdx.x == 0 && blockIdx.x == 0) {
    float s0 = 0.f, s1 = 0.f;
    for (int f = 0; f < F; ++f) {
      s0 += q[f] * partial[f] * invn;
      s1 += q[f] * partial[F + f] * invn;
    }
    float mx = fmaxf(s0, s1);
    float e0 = __expf(s0 - mx), e1 = __expf(s1 - mx);
    float den = e0 + e1;
    attn[0] = e0 / den;
    attn[1] = e1 / den;
  }
}

// out[n,f] = attn0*relu(m0[n,f]) + attn1*relu(m1[n,f])
template <int F>
__global__ void combine2_kernel(const float* __restrict__ m0, const float* __restrict__ m1,
                                int ms, const float* __restrict__ attn,
                                float* __restrict__ out, int os, int n) {
  int i = blockIdx.x * blockDim.x + threadIdx.x;
  if (i >= n) return;
  float a0 = attn[0], a1 = attn[1];
#pragma unroll
  for (int f = 0; f < F; ++f)
    out[(size_t)i * os + f] =
        a0 * reluf(m0[(size_t)i * ms + f]) + a1 * reluf(m1[(size_t)i * ms + f]);
}

template <int F>
__global__ void relu_copy_kernel(const float* __restrict__ in, int is,
                                 float* __restrict__ out, int os, int n) {
  int i = blockIdx.x * blockDim.x + threadIdx.x;
  if (i >= n) return;
#pragma unroll
  for (int f = 0; f < F; ++f) out[(size_t)i * os + f] = reluf(in[(size_t)i * is + f]);
}

__global__ void log_softmax7_kernel(const float* __restrict__ x, int xs,
                                    float* __restrict__ out, int n) {
  int i = blockIdx.x * blockDim.x + threadIdx.x;
  if (i >= n) return;
  float v[7];
  float mx = -3.0e38f;
#pragma unroll
  for (int f = 0; f < 7; ++f) { v[f] = x[(size_t)i * xs + f]; mx = fmaxf(mx, v[f]); }
  float s = 0.f;
#pragma unroll
  for (int f = 0; f < 7; ++f) s += __expf(v[f] - mx);
  float lse = __logf(s);
#pragma unroll
  for (int f = 0; f < 7; ++f) out[(size_t)i * 7 + f] = v[f] - mx - lse;
}

// ---------------------------------------------------------------------------
// launch
// ---------------------------------------------------------------------------
extern "C" void kernel_launch(void* const* d_in, const int* in_sizes, int n_in,
                              void* d_out, int out_size, void* d_ws, size_t ws_size,
                              hipStream_t stream) {
  const int NP = NP_N, NA = NA_N, E = E_N;
  const float NEG = -3.0e38f;

  // ---- inputs (setup_inputs dict order) ----
  const float* x_p      = (const float*)d_in[0];
  const float* x_a      = (const float*)d_in[1];
  const int*   ei_pp    = (const int*)d_in[2];
  const int*   ei_pa    = (const int*)d_in[3];
  const int*   ei_ap    = (const int*)d_in[4];
  const float* w1_p     = (const float*)d_in[5];
  const float* b1_p     = (const float*)d_in[6];
  const float* w1_a     = (const float*)d_in[7];
  const float* b1_a     = (const float*)d_in[8];
  const float* asrc1_pp = (const float*)d_in[9];
  const float* adst1_pp = (const float*)d_in[10];
  const float* asrc1_pa = (const float*)d_in[11];
  const float* adst1_pa = (const float*)d_in[12];
  const float* asrc1_ap = (const float*)d_in[13];
  const float* adst1_ap = (const float*)d_in[14];
  const float* kw1      = (const float*)d_in[15];
  const float* kb1      = (const float*)d_in[16];
  const float* q1       = (const float*)d_in[17];
  const float* w2_p     = (const float*)d_in[18];
  const float* b2_p     = (const float*)d_in[19];
  const float* w2_a     = (const float*)d_in[20];
  const float* b2_a     = (const float*)d_in[21];
  const float* asrc2_pp = (const float*)d_in[22];
  const float* adst2_pp = (const float*)d_in[23];
  // d_in[24], d_in[25]: asrc2_pa/adst2_pa (author output unused in layer 2)
  const float* asrc2_ap = (const float*)d_in[26];
  const float* adst2_ap = (const float*)d_in[27];
  const float* kw2      = (const float*)d_in[28];
  const float* kb2      = (const float*)d_in[29];
  const float* q2       = (const float*)d_in[30];
  float* out = (float*)d_out;

  // ---- workspace layout (floats) ----
  float* ws = (float*)d_ws;
  size_t off = 0;
  auto A = [&](size_t n) { float* p = ws + off; off += n; return p; };
  float* xp_p   = A((size_t)NP * 8);  float* xp_a   = A((size_t)NA * 8);
  float* as_pp  = A((size_t)NP * 2);  float* ad_pp  = A((size_t)NP * 2);
  float* as_pa  = A((size_t)NP * 2);  float* ad_pa  = A((size_t)NA * 2);
  float* as_ap  = A((size_t)NA * 2);  float* ad_ap  = A((size_t)NP * 2);
  float* m_pp   = A((size_t)NP * 2);  float* s_pp   = A((size_t)NP * 2);
  float* m_pa   = A((size_t)NA * 2);  float* s_pa   = A((size_t)NA * 2);
  float* m_ap   = A((size_t)NP * 2);  float* s_ap   = A((size_t)NP * 2);
  float* msg_pp = A((size_t)NP * 8);  float* msg_pa = A((size_t)NA * 8);
  float* msg_ap = A((size_t)NP * 8);
  float* part1  = A(16);              float* attn1  = A(2);
  float* x1_p   = A((size_t)NP * 8);  float* x1_a   = A((size_t)NA * 8);
  float* xp2_p  = A((size_t)NP * 8);  float* xp2_a  = A((size_t)NA * 8);
  float* as2_pp = A((size_t)NP);      float* ad2_pp = A((size_t)NP);
  float* as2_ap = A((size_t)NA);      float* ad2_ap = A((size_t)NP);
  float* m2_pp  = A((size_t)NP);      float* s2_pp  = A((size_t)NP);
  float* m2_ap  = A((size_t)NP);      float* s2_ap  = A((size_t)NP);
  float* msg2_pp = A((size_t)NP * 8); float* msg2_ap = A((size_t)NP * 8);
  float* part2  = A(16);              float* attn2  = A(2);
  float* x2_p   = A((size_t)NP * 8);
  size_t total = off;
  (void)ws_size; (void)n_in; (void)in_sizes; (void)out_size;

  const int TB = 256;
  int nb_node = cdiv(NP, TB);      // NP == NA
  int nb_edge = cdiv(E, TB);
  int nb_fill = (int)((total + TB - 1) / TB);

  // ---- init: zero everything we use, then -inf the max buffers ----
  fill_kernel<<<nb_fill, TB, 0, stream>>>(ws, 0.f, total);
  fill_kernel<<<cdiv(NP * 2, TB), TB, 0, stream>>>(m_pp, NEG, (size_t)NP * 2);
  fill_kernel<<<cdiv(NA * 2, TB), TB, 0, stream>>>(m_pa, NEG, (size_t)NA * 2);
  fill_kernel<<<cdiv(NP * 2, TB), TB, 0, stream>>>(m_ap, NEG, (size_t)NP * 2);
  fill_kernel<<<cdiv(NP, TB), TB, 0, stream>>>(m2_pp, NEG, (size_t)NP);
  fill_kernel<<<cdiv(NP, TB), TB, 0, stream>>>(m2_ap, NEG, (size_t)NP);

  // ================== LAYER 1 (H=2, D=4, F=8) ==================
  {
    int tiles = cdiv(NP, 16);
    int nb = cdiv(tiles, TB / 32);
    wmma_proj_kernel<<<nb, TB, 0, stream>>>(x_p, w1_p, b1_p, xp_p, NP, 128, 8, 8);
    wmma_proj_kernel<<<nb, TB, 0, stream>>>(x_a, w1_a, b1_a, xp_a, NA, 128, 8, 8);
  }
  alpha_kernel<2, 4><<<nb_node, TB, 0, stream>>>(xp_p, 8, asrc1_pp, as_pp, NP);
  alpha_kernel<2, 4><<<nb_node, TB, 0, stream>>>(xp_p, 8, adst1_pp, ad_pp, NP);
  alpha_kernel<2, 4><<<nb_node, TB, 0, stream>>>(xp_p, 8, asrc1_pa, as_pa, NP);
  alpha_kernel<2, 4><<<nb_node, TB, 0, stream>>>(xp_a, 8, adst1_pa, ad_pa, NA);
  alpha_kernel<2, 4><<<nb_node, TB, 0, stream>>>(xp_a, 8, asrc1_ap, as_ap, NA);
  alpha_kernel<2, 4><<<nb_node, TB, 0, stream>>>(xp_p, 8, adst1_ap, ad_ap, NP);

  // paper -> paper
  edge_max_kernel<2><<<nb_edge, TB, 0, stream>>>(ei_pp, ei_pp + E, as_pp, ad_pp, m_pp, E);
  edge_sumexp_kernel<2><<<nb_edge, TB, 0, stream>>>(ei_pp, ei_pp + E, as_pp, ad_pp, m_pp, s_pp, E);
  edge_msg_kernel<2, 4><<<nb_edge, TB, 0, stream>>>(ei_pp, ei_pp + E, as_pp, ad_pp, m_pp, s_pp,
                                                    xp_p, 8, msg_pp, 8, E);
  // paper -> author
  edge_max_kernel<2><<<nb_edge, TB, 0, stream>>>(ei_pa, ei_pa + E, as_pa, ad_pa, m_pa, E);
  edge_sumexp_kernel<2><<<nb_edge, TB, 0, stream>>>(ei_pa, ei_pa + E, as_pa, ad_pa, m_pa, s_pa, E);
  edge_msg_kernel<2, 4><<<nb_edge, TB, 0, stream>>>(ei_pa, ei_pa + E, as_pa, ad_pa, m_pa, s_pa,
                                                    xp_p, 8, msg_pa, 8, E);
  // author -> paper
  edge_max_kernel<2><<<nb_edge, TB, 0, stream>>>(ei_ap, ei_ap + E, as_ap, ad_ap, m_ap, E);
  edge_sumexp_kernel<2><<<nb_edge, TB, 0, stream>>>(ei_ap, ei_ap + E, as_ap, ad_ap, m_ap, s_ap, E);
  edge_msg_kernel<2, 4><<<nb_edge, TB, 0, stream>>>(ei_ap, ei_ap + E, as_ap, ad_ap, m_ap, s_ap,
                                                    xp_a, 8, msg_ap, 8, E);

  // semantic attention over {pp, ap} for paper; author has single edge type (pa)
  sem_reduce_kernel<8><<<nb_node, TB, 0, stream>>>(msg_pp, 8, kw1, kb1, part1, NP);
  sem_reduce_kernel<8><<<nb_node, TB, 0, stream>>>(msg_ap, 8, kw1, kb1, part1 + 8, NP);
  sem_attn_kernel<<<1, 32, 0, stream>>>(part1, q1, 8, 1.0f / NP, attn1);
  combine2_kernel<8><<<nb_node, TB, 0, stream>>>(msg_pp, msg_ap, 8, attn1, x1_p, 8, NP);
  relu_copy_kernel<8><<<nb_node, TB, 0, stream>>>(msg_pa, 8, x1_a, 8, NA);
  // inter-layer ReLU is a no-op (outputs already non-negative)

  // ================== LAYER 2 (H=1, D=7, F=7) ==================
  {
    int tiles = cdiv(NP, 16);
    int nb = cdiv(tiles, TB / 32);
    wmma_proj_kernel<<<nb, TB, 0, stream>>>(x1_p, w2_p, b2_p, xp2_p, NP, 8, 7, 8);
    wmma_proj_kernel<<<nb, TB, 0, stream>>>(x1_a, w2_a, b2_a, xp2_a, NA, 8, 7, 8);
  }
  alpha_kernel<1, 7><<<nb_node, TB, 0, stream>>>(xp2_p, 8, asrc2_pp, as2_pp, NP);
  alpha_kernel<1, 7><<<nb_node, TB, 0, stream>>>(xp2_p, 8, adst2_pp, ad2_pp, NP);
  alpha_kernel<1, 7><<<nb_node, TB, 0, stream>>>(xp2_a, 8, asrc2_ap, as2_ap, NA);
  alpha_kernel<1, 7><<<nb_node, TB, 0, stream>>>(xp2_p, 8, adst2_ap, ad2_ap, NP);

  // only edge types reaching "paper" matter for the final output
  edge_max_kernel<1><<<nb_edge, TB, 0, stream>>>(ei_pp, ei_pp + E, as2_pp, ad2_pp, m2_pp, E);
  edge_sumexp_kernel<1><<<nb_edge, TB, 0, stream>>>(ei_pp, ei_pp + E, as2_pp, ad2_pp, m2_pp, s2_pp, E);
  edge_msg_kernel<1, 7><<<nb_edge, TB, 0, stream>>>(ei_pp, ei_pp + E, as2_pp, ad2_pp, m2_pp, s2_pp,
                                                    xp2_p, 8, msg2_pp, 8, E);
  edge_max_kernel<1><<<nb_edge, TB, 0, stream>>>(ei_ap, ei_ap + E, as2_ap, ad2_ap, m2_ap, E);
  edge_sumexp_kernel<1><<<nb_edge, TB, 0, stream>>>(ei_ap, ei_ap + E, as2_ap, ad2_ap, m2_ap, s2_ap, E);
  edge_msg_kernel<1, 7><<<nb_edge, TB, 0, stream>>>(ei_ap, ei_ap + E, as2_ap, ad2_ap, m2_ap, s2_ap,
                                                    xp2_a, 8, msg2_ap, 8, E);

  sem_reduce_kernel<7><<<nb_node, TB, 0, stream>>>(msg2_pp, 8, kw2, kb2, part2, NP);
  sem_reduce_kernel<7><<<nb_node, TB, 0, stream>>>(msg2_ap, 8, kw2, kb2, part2 + 7, NP);
  sem_attn_kernel<<<1, 32, 0, stream>>>(part2, q2, 7, 1.0f / NP, attn2);
  combine2_kernel<7><<<nb_node, TB, 0, stream>>>(msg2_pp, msg2_ap, 8, attn2, x2_p, 8, NP);

  log_softmax7_kernel<<<nb_node, TB, 0, stream>>>(x2_p, 8, out, NP);
}